// SennLocalLayer_9990093930938
// MI455X (gfx1250) — compile-verified
//
#include <hip/hip_runtime.h>

#define B_ROWS 8192
#define IN_DIM 4096
#define OUT_DIM 1024
#define NCHUNK 64
#define ROWS_PER_CHUNK (B_ROWS / NCHUNK) /* 128 */
#define KCHUNKS 32                        /* K split for matvec */

typedef __attribute__((ext_vector_type(2))) float v2f;
typedef __attribute__((ext_vector_type(8))) float v8f;

struct alignas(16) F4 { float x, y, z, w; };

// ---------------------------------------------------------------------------
// K1: partial column-sum of x : [8192, 4096] -> partials[NCHUNK][4096]
// Coalesced float4 (global_load_b128) streaming; this is the 128 MB pass that
// dominates the roofline, so it gets prefetch + wide loads.
// ---------------------------------------------------------------------------
__global__ void colsum_partial(const float* __restrict__ x,
                               float* __restrict__ partials) {
    const int cols4 = IN_DIM / 4;                         // 1024
    int c4 = blockIdx.x * blockDim.x + threadIdx.x;       // 0..1023
    int chunk = blockIdx.y;                               // 0..63
    int r0 = chunk * ROWS_PER_CHUNK;
    const F4* x4 = (const F4*)x;
    F4 acc; acc.x = acc.y = acc.z = acc.w = 0.0f;
    for (int r = 0; r < ROWS_PER_CHUNK; ++r) {
        const F4* p = &x4[(size_t)(r0 + r) * cols4 + c4];
        if (r + 8 < ROWS_PER_CHUNK)
            __builtin_prefetch((const void*)(p + (size_t)8 * cols4), 0, 3);
        F4 val = *p;
        acc.x += val.x; acc.y += val.y; acc.z += val.z; acc.w += val.w;
    }
    ((F4*)partials)[(size_t)chunk * cols4 + c4] = acc;
}

// ---------------------------------------------------------------------------
// K2: xsum[i] = sum over chunks of partials[c][i]   (deterministic order)
// ---------------------------------------------------------------------------
__global__ void reduce_partials(const float* __restrict__ partials,
                                float* __restrict__ xsum) {
    int i = blockIdx.x * blockDim.x + threadIdx.x;        // 0..4095
    float s = 0.0f;
    for (int c = 0; c < NCHUNK; ++c) s += partials[(size_t)c * IN_DIM + i];
    xsum[i] = s;
}

// ---------------------------------------------------------------------------
// K3: y = xsum @ W via V_WMMA_F32_16X16X4_F32 (exact f32 path).
// A is 16x4 with the xsum slice in row M=0, zeros elsewhere; B is
// W[k:k+4, n0:n0+16]. A layout (32-bit 16x4): lane = M (mod 16); lanes 0-15
// hold K=0/1 in v[0]/v[1], lanes 16-31 hold K=2/3. D row M=0 is VGPR0,
// lanes 0-15. One wave per (N-tile, K-chunk).
//
// The wave's 128 xsum values are loaded ONCE (one b128 per lane, coalesced)
// and broadcast per-iteration with __shfl, so the unrolled inner loop has
// only the two W loads + the WMMA — no predicated loads, no exec toggling.
// ---------------------------------------------------------------------------
__global__ void matvec_wmma(const float* __restrict__ xsum,
                            const float* __restrict__ W,
                            float* __restrict__ ypart) {
    int lane = threadIdx.x;       // 0..31 (full wave -> EXEC all ones)
    int half = lane >> 4;         // selects K pair {0,1} vs {2,3}
    int ln   = lane & 15;
    int n0   = blockIdx.x * 16;   // 64 N tiles
    int kc   = blockIdx.y * (IN_DIM / KCHUNKS);  // 32 chunks of 128

    // lane L holds xsum[kc + 4L .. 4L+3]
    F4 xs = ((const F4*)xsum)[kc / 4 + lane];

    const float* wp = W + (size_t)(kc + half * 2) * OUT_DIM + n0 + ln;
    v8f c = {};
    #pragma unroll
    for (int t = 0; t < IN_DIM / KCHUNKS / 4; ++t) {      // 32 WMMAs
        float sx = __shfl(xs.x, t, 32);
        float sy = __shfl(xs.y, t, 32);
        float sz = __shfl(xs.z, t, 32);
        float sw = __shfl(xs.w, t, 32);
        v2f a;
        a.x = (ln == 0) ? (half ? sz : sx) : 0.0f;
        a.y = (ln == 0) ? (half ? sw : sy) : 0.0f;
        v2f b;
        b.x = wp[0];
        b.y = wp[OUT_DIM];
        wp += 4 * OUT_DIM;
        c = __builtin_amdgcn_wmma_f32_16x16x4_f32(
                false, a, false, b, (short)0, c, false, false);
    }
    if (lane < 16)                // D: M=0 is VGPR0, lanes 0-15 (N=lane)
        ypart[(size_t)blockIdx.y * OUT_DIM + n0 + lane] = c[0];
}

// ---------------------------------------------------------------------------
// Threefry-2x32-20 (JAX's generator; key(42) -> (0,42); counts = iota(1024)
// split in halves).
// ---------------------------------------------------------------------------
__device__ __forceinline__ unsigned rotl32(unsigned x, int r) {
    return (x << r) | (x >> (32 - r));
}

__device__ void threefry2x32_20(unsigned k0, unsigned k1, unsigned c0,
                                unsigned c1, unsigned& o0, unsigned& o1) {
    unsigned ks0 = k0, ks1 = k1, ks2 = 0x1BD11BDAu ^ k0 ^ k1;
    unsigned x0 = c0 + ks0, x1 = c1 + ks1;
    const int ra[4] = {13, 15, 26, 6};
    const int rb[4] = {17, 29, 16, 24};
    #pragma unroll
    for (int r = 0; r < 4; ++r) { x0 += x1; x1 = rotl32(x1, ra[r]); x1 ^= x0; }
    x0 += ks1; x1 += ks2 + 1u;
    #pragma unroll
    for (int r = 0; r < 4; ++r) { x0 += x1; x1 = rotl32(x1, rb[r]); x1 ^= x0; }
    x0 += ks2; x1 += ks0 + 2u;
    #pragma unroll
    for (int r = 0; r < 4; ++r) { x0 += x1; x1 = rotl32(x1, ra[r]); x1 ^= x0; }
    x0 += ks0; x1 += ks1 + 3u;
    #pragma unroll
    for (int r = 0; r < 4; ++r) { x0 += x1; x1 = rotl32(x1, rb[r]); x1 ^= x0; }
    x0 += ks1; x1 += ks2 + 4u;
    #pragma unroll
    for (int r = 0; r < 4; ++r) { x0 += x1; x1 = rotl32(x1, ra[r]); x1 ^= x0; }
    x0 += ks2; x1 += ks0 + 5u;
    o0 = x0; o1 = x1;
}

// ---------------------------------------------------------------------------
// K4: reduce ypart, membrane EMA, sigmoid, bernoulli, delta.
// Writes z / rho / v_new to d_out[0..3072), delta to ws.
// ---------------------------------------------------------------------------
__global__ void finalize_small(const float* __restrict__ ypart,
                               const float* __restrict__ v,
                               float* __restrict__ out,
                               float* __restrict__ delta_ws) {
    int j = blockIdx.x * blockDim.x + threadIdx.x;        // 0..1023
    float y = 0.0f;
    for (int c = 0; c < KCHUNKS; ++c) y += ypart[(size_t)c * OUT_DIM + j];
    float imean = y * (1.0f / (float)B_ROWS);
    float vn  = v[j] * (1.0f - 1.0f / 20.0f) + imean;     // TAU_MEM = 20
    float rho = 1.0f / (1.0f + __expf(-vn));
    unsigned i = (unsigned)(j & 511);
    unsigned o0, o1;
    threefry2x32_20(0u, 42u, i, i + 512u, o0, o1);
    unsigned bits = (j < 512) ? o0 : o1;
    float u = __uint_as_float((bits >> 9) | 0x3f800000u) - 1.0f;
    float z = (u < rho) ? 1.0f : 0.0f;
    out[j]               = z;
    out[OUT_DIM + j]     = rho;
    out[2 * OUT_DIM + j] = vn;
    delta_ws[j] = z - rho;
}

// ---------------------------------------------------------------------------
// K5: fused eligibility + weight update, float4 loads/stores (64 MB traffic).
// elig_new = 0.99*elig + xsum[i]*delta[j];  W_new = W + (LR/B)*elig_new
// ---------------------------------------------------------------------------
__global__ void update_elig_w(const float* __restrict__ elig,
                              const float* __restrict__ W,
                              const float* __restrict__ xsum,
                              const float* __restrict__ delta,
                              float* __restrict__ elig_out,
                              float* __restrict__ w_out) {
    size_t t = (size_t)blockIdx.x * blockDim.x + threadIdx.x; // 0..2^20-1
    int i  = (int)(t >> 8);     // input row 0..4095
    int c4 = (int)(t & 255);    // float4 column 0..255
    F4 e = ((const F4*)elig)[t];
    F4 w = ((const F4*)W)[t];
    F4 d = ((const F4*)delta)[c4];
    float xs = xsum[i];
    const float ke = 1.0f - 1.0f / 100.0f;   // TAU_E = 100
    const float kw = 0.001f / (float)B_ROWS; // LR / B
    F4 en, wn;
    en.x = ke * e.x + xs * d.x;  wn.x = w.x + kw * en.x;
    en.y = ke * e.y + xs * d.y;  wn.y = w.y + kw * en.y;
    en.z = ke * e.z + xs * d.z;  wn.z = w.z + kw * en.z;
    en.w = ke * e.w + xs * d.w;  wn.w = w.w + kw * en.w;
    ((F4*)elig_out)[t] = en;
    ((F4*)w_out)[t]    = wn;
}

extern "C" void kernel_launch(void* const* d_in, const int* in_sizes, int n_in,
                              void* d_out, int out_size, void* d_ws, size_t ws_size,
                              hipStream_t stream) {
    (void)in_sizes; (void)n_in; (void)out_size; (void)ws_size;
    const float* x    = (const float*)d_in[0];   // [8192, 4096]
    const float* W    = (const float*)d_in[1];   // [4096, 1024]
    const float* v    = (const float*)d_in[2];   // [1024]
    const float* elig = (const float*)d_in[3];   // [4096, 1024]
    float* out = (float*)d_out;

    // ws layout (floats): partials[64*4096] | xsum[4096] | ypart[32*1024] | delta[1024]
    float* ws       = (float*)d_ws;
    float* partials = ws;
    float* xsum     = partials + (size_t)NCHUNK * IN_DIM;
    float* ypart    = xsum + IN_DIM;
    float* delta    = ypart + (size_t)KCHUNKS * OUT_DIM;

    colsum_partial<<<dim3((IN_DIM / 4) / 256, NCHUNK), 256, 0, stream>>>(x, partials);
    reduce_partials<<<IN_DIM / 256, 256, 0, stream>>>(partials, xsum);
    matvec_wmma<<<dim3(OUT_DIM / 16, KCHUNKS), 32, 0, stream>>>(xsum, W, ypart);
    finalize_small<<<OUT_DIM / 256, 256, 0, stream>>>(ypart, v, out, delta);

    float* elig_out = out + 3 * OUT_DIM;
    float* w_out    = elig_out + (size_t)IN_DIM * OUT_DIM;
    update_elig_w<<<((size_t)IN_DIM * OUT_DIM / 4) / 256, 256, 0, stream>>>(
        elig, W, xsum, delta, elig_out, w_out);
}